// GLR_43267500540083
// MI455X (gfx1250) — compile-verified
//
#include <hip/hip_runtime.h>
#include <cstddef>
#include <cstdint>

// ---------------------------------------------------------------------------
// GLR forward for MI455X (gfx1250, wave32).
//  - CNN stages: plain VALU kernels (~5 GFLOP, negligible).
//  - Dominant cost: 16x inversion of 1296x1296 SPD A = I + u*L  (~70 GFLOP).
//    Blocked in-place Gauss-Jordan over 81x81 grid of 16x16 blocks, all block
//    GEMMs via V_WMMA_F32_16X16X4_F32 (4 chained K=4 WMMAs per 16x16x16).
//    A (107 MB for all 16 batches) lives in the 192 MB global L2.
//  - Elimination kernel: 4x4 block tile per workgroup; pivot-row tile staged
//    in LDS via GLOBAL_LOAD_ASYNC_TO_LDS_B128 + s_wait_asynccnt (builtins
//    confirmed present; plain staging fallback kept for portability); Col
//    block kept in registers as WMMA A-fragments, reused across 4 j-tiles.
//  - Final A^{-1} @ y: y (1296x3) zero-padded to 1296x16, WMMA GEMM.
// Workspace requirement: ~145 MB.
// ---------------------------------------------------------------------------

static constexpr int BATCH  = 16;
static constexpr int WID    = 36;
static constexpr int NNODE  = WID * WID;   // 1296 (= 81 * 16, WMMA friendly)
static constexpr int NBLK   = NNODE / 16;  // 81

#define ACT_NONE  0
#define ACT_RELU  1
#define ACT_LRELU 2

// Async LDS staging only if this toolchain exposes both builtins.
#if defined(__HIP_DEVICE_COMPILE__) && \
    __has_builtin(__builtin_amdgcn_global_load_async_to_lds_b128) && \
    __has_builtin(__builtin_amdgcn_s_wait_asynccnt)
#define USE_ASYNC_LDS 1
#else
#define USE_ASYNC_LDS 0
#endif

typedef float v2f __attribute__((ext_vector_type(2)));
typedef float v8f __attribute__((ext_vector_type(8)));

#if USE_ASYNC_LDS
// Exact parameter types per the compiler diagnostic:
//   (int __attribute__((vector_size(16))) AS(1)*,  ... AS(3)*, Ii, Ii)
typedef int v4i_vs __attribute__((vector_size(4 * sizeof(int))));
typedef __attribute__((address_space(1))) v4i_vs* gv4i_ptr;
typedef __attribute__((address_space(3))) v4i_vs* lv4i_ptr;
#endif

// ---------------------------- WMMA helpers ---------------------------------
// V_WMMA_F32_16X16X4_F32 operand layouts (ISA 7.12.2, wave32):
//  A 16x4 : lanes 0-15 hold M=lane (K = 0,1 in the 2 VGPRs),
//           lanes 16-31 hold M=lane-16 (K = 2,3).
//  B 4x16 : rows striped across lanes within a VGPR (mirror of A).
//  C/D 16x16 f32: VGPR r -> row r (lanes 0-15) / row r+8 (lanes 16-31).
__device__ __forceinline__ v8f v8f_zero() {
  v8f z;
#pragma unroll
  for (int r = 0; r < 8; ++r) z[r] = 0.0f;
  return z;
}

struct AFrag { v2f a[4]; };  // 16x16 A-operand, pre-striped for 4 K=4 steps

__device__ __forceinline__ AFrag load_afrag(const float* __restrict__ Ab, int lda) {
  const int lane = threadIdx.x & 31;
  const int half = lane >> 4;
  const int l15  = lane & 15;
  AFrag f;
#pragma unroll
  for (int t = 0; t < 4; ++t) {
    const int ka = t * 4 + half * 2;
    f.a[t].x = Ab[l15 * lda + ka];
    f.a[t].y = Ab[l15 * lda + ka + 1];
  }
  return f;
}

__device__ __forceinline__ v8f wmma_afrag(const AFrag& f, const float* __restrict__ Bb,
                                          int ldb, v8f acc) {
  const int lane = threadIdx.x & 31;
  const int half = lane >> 4;
  const int l15  = lane & 15;
#pragma unroll
  for (int t = 0; t < 4; ++t) {
    const int ka = t * 4 + half * 2;
    v2f b;
    b.x = Bb[(ka)     * ldb + l15];
    b.y = Bb[(ka + 1) * ldb + l15];
    // 8 args: (neg_a, A, neg_b, B, c_mod, C, reuse_a, reuse_b); NEG=0 for f32.
    acc = __builtin_amdgcn_wmma_f32_16x16x4_f32(false, f.a[t], false, b,
                                                (short)0, acc, false, false);
  }
  return acc;
}

// acc += Ablk(16x16, lda) * Bblk(16x16, ldb), one wave.
__device__ __forceinline__ v8f wmma_blk16(const float* __restrict__ Ab, int lda,
                                          const float* __restrict__ Bb, int ldb,
                                          v8f acc) {
  return wmma_afrag(load_afrag(Ab, lda), Bb, ldb, acc);
}

__device__ __forceinline__ v8f load_c16(const float* __restrict__ C, int ldc) {
  const int lane = threadIdx.x & 31;
  const int half = lane >> 4;
  const int n    = lane & 15;
  v8f c;
#pragma unroll
  for (int r = 0; r < 8; ++r) c[r] = C[(r + 8 * half) * ldc + n];
  return c;
}

__device__ __forceinline__ void store_c16(float* __restrict__ D, int ldd, v8f c) {
  const int lane = threadIdx.x & 31;
  const int half = lane >> 4;
  const int n    = lane & 15;
#pragma unroll
  for (int r = 0; r < 8; ++r) D[(r + 8 * half) * ldd + n] = c[r];
}

// ---------------------------- CNN kernels ----------------------------------

__device__ __forceinline__ float act_apply(float v, int act) {
  if (act == ACT_RELU)  return v > 0.0f ? v : 0.0f;
  if (act == ACT_LRELU) return v >= 0.0f ? v : 0.05f * v;
  return v;
}

// 3x3 conv, pad 1, arbitrary stride; output written into a (possibly larger)
// channel-concatenated buffer [B, ctot, Ho, Wo] at channel offset coff.
__global__ void conv3x3_kernel(const float* __restrict__ in,
                               const float* __restrict__ w,
                               const float* __restrict__ bias,
                               float* __restrict__ out,
                               int Cin, int H, int W, int Cout,
                               int stride, int Ho, int Wo,
                               int act, int ctot, int coff) {
  int idx = blockIdx.x * blockDim.x + threadIdx.x;
  int total = BATCH * Cout * Ho * Wo;
  if (idx >= total) return;
  int wo = idx % Wo;
  int ho = (idx / Wo) % Ho;
  int co = (idx / (Wo * Ho)) % Cout;
  int b  = idx / (Wo * Ho * Cout);
  float acc = bias[co];
  const float* inb = in + (size_t)b * Cin * H * W;
  const float* wc  = w + (size_t)co * Cin * 9;
  for (int ci = 0; ci < Cin; ++ci) {
    const float* ip = inb + (size_t)ci * H * W;
    const float* wp = wc + ci * 9;
#pragma unroll
    for (int dy = 0; dy < 3; ++dy) {
      int iy = ho * stride + dy - 1;
      if (iy < 0 || iy >= H) continue;
#pragma unroll
      for (int dx = 0; dx < 3; ++dx) {
        int ix = wo * stride + dx - 1;
        if (ix < 0 || ix >= W) continue;
        acc += ip[iy * W + ix] * wp[dy * 3 + dx];
      }
    }
  }
  out[(((size_t)b * ctot + coff + co) * Ho + ho) * Wo + wo] = act_apply(acc, act);
}

// reflect-pad(1) coordinate for an upsampled(2x) image of size n2, then map
// back to source pixel (jnp 'reflect': [-1] -> 1, [n2] -> n2-2).
__device__ __forceinline__ int refl_up(int p, int n2) {
  int u = p - 1;
  if (u < 0) u = -u;
  if (u >= n2) u = 2 * n2 - 2 - u;
  return u >> 1;
}

// conv3x3(pad=0) over refpad1(upsample2(in)):  in [B,Cin,H,W] -> out 2H x 2W
__global__ void upconv_kernel(const float* __restrict__ in,
                              const float* __restrict__ w,
                              const float* __restrict__ bias,
                              float* __restrict__ out,
                              int Cin, int H, int W, int Cout,
                              int ctot, int coff) {
  int Ho = 2 * H, Wo = 2 * W;
  int idx = blockIdx.x * blockDim.x + threadIdx.x;
  int total = BATCH * Cout * Ho * Wo;
  if (idx >= total) return;
  int wo = idx % Wo;
  int ho = (idx / Wo) % Ho;
  int co = (idx / (Wo * Ho)) % Cout;
  int b  = idx / (Wo * Ho * Cout);
  float acc = bias[co];
  const float* inb = in + (size_t)b * Cin * H * W;
  const float* wc  = w + (size_t)co * Cin * 9;
  int iy[3], ix[3];
#pragma unroll
  for (int d = 0; d < 3; ++d) {
    iy[d] = refl_up(ho + d, Ho);
    ix[d] = refl_up(wo + d, Wo);
  }
  for (int ci = 0; ci < Cin; ++ci) {
    const float* ip = inb + (size_t)ci * H * W;
    const float* wp = wc + ci * 9;
#pragma unroll
    for (int dy = 0; dy < 3; ++dy)
#pragma unroll
      for (int dx = 0; dx < 3; ++dx)
        acc += ip[iy[dy] * W + ix[dx]] * wp[dy * 3 + dx];
  }
  out[(((size_t)b * ctot + coff + co) * Ho + ho) * Wo + wo] = acc;
}

// 2x2 maxpool, stride 2, ceil mode (bounds-guarded == -inf padding)
__global__ void maxpool_kernel(const float* __restrict__ in, float* __restrict__ out,
                               int C, int H, int W) {
  int Ho = (H + 1) / 2, Wo = (W + 1) / 2;
  int idx = blockIdx.x * blockDim.x + threadIdx.x;
  int total = BATCH * C * Ho * Wo;
  if (idx >= total) return;
  int wo = idx % Wo;
  int ho = (idx / Wo) % Ho;
  int c  = (idx / (Wo * Ho)) % C;
  int b  = idx / (Wo * Ho * C);
  const float* ip = in + (((size_t)b * C + c) * H) * W;
  float m = -INFINITY;
#pragma unroll
  for (int dy = 0; dy < 2; ++dy) {
    int y = 2 * ho + dy;
    if (y >= H) continue;
#pragma unroll
    for (int dx = 0; dx < 2; ++dx) {
      int x = 2 * wo + dx;
      if (x >= W) continue;
      m = fmaxf(m, ip[y * W + x]);
    }
  }
  out[(((size_t)b * C + c) * Ho + ho) * Wo + wo] = m;
}

// copy [B,C,HW] into channel-offset slice of [B,ctot,HW] (for concat)
__global__ void copych_kernel(const float* __restrict__ src, float* __restrict__ dst,
                              int C, int HW, int ctot, int coff) {
  int idx = blockIdx.x * blockDim.x + threadIdx.x;
  int total = BATCH * C * HW;
  if (idx >= total) return;
  int p = idx % HW;
  int c = (idx / HW) % C;
  int b = idx / (HW * C);
  dst[((size_t)b * ctot + coff + c) * HW + p] = src[((size_t)b * C + c) * HW + p];
}

__global__ void resadd_kernel(const float* __restrict__ x, const float* __restrict__ t,
                              float* __restrict__ y, int n) {
  int i = blockIdx.x * blockDim.x + threadIdx.x;
  if (i < n) y[i] = x[i] + t[i];
}

// cnnu tail: flatten [B,32,3,3] -> fc(288->32) -> fc(32->1) -> relu -> clip
__global__ void cnnu_fc_kernel(const float* __restrict__ in,
                               const float* __restrict__ w1, const float* __restrict__ b1,
                               const float* __restrict__ w2, const float* __restrict__ b2,
                               float* __restrict__ u) {
  int b = blockIdx.x;
  int j = threadIdx.x;  // 32 threads
  __shared__ float h[32];
  const float* xb = in + (size_t)b * 288;
  float acc = b1[j];
  for (int i = 0; i < 288; ++i) acc += w1[j * 288 + i] * xb[i];
  h[j] = acc;
  __syncthreads();
  if (j == 0) {
    float o = b2[0];
#pragma unroll
    for (int t = 0; t < 32; ++t) o += w2[t] * h[t];
    o = fmaxf(o, 0.0f);                       // relu
    o = fminf(fmaxf(o, 0.001f), 15.5625f);    // clip
    u[b] = o;
  }
}

// ---------------------------- Graph / solve --------------------------------

__global__ void zero_kernel(float* __restrict__ p, long long n) {
  long long i = (long long)blockIdx.x * blockDim.x + threadIdx.x;
  long long stride = (long long)gridDim.x * blockDim.x;
  for (; i < n; i += stride) p[i] = 0.0f;
}

// Build A = I + u*(D - W) row-by-row: each thread fills one row's <=9 nnz
// over the 8-connected 36x36 grid.  w_ij = exp(-||E_i - E_j||^2 / 2).
__global__ void buildA_kernel(const float* __restrict__ E, const float* __restrict__ u,
                              float* __restrict__ A) {
  int idx = blockIdx.x * blockDim.x + threadIdx.x;
  if (idx >= BATCH * NNODE) return;
  int b = idx / NNODE;
  int i = idx % NNODE;
  int y = i / WID, x = i % WID;
  const float ub = u[b];
  const float* Eb = E + (size_t)b * 3 * NNODE;
  float e0 = Eb[i], e1 = Eb[NNODE + i], e2 = Eb[2 * NNODE + i];
  float deg = 0.0f;
  float* row = A + (size_t)b * NNODE * NNODE + (size_t)i * NNODE;
#pragma unroll
  for (int dy = -1; dy <= 1; ++dy) {
#pragma unroll
    for (int dx = -1; dx <= 1; ++dx) {
      if (dy == 0 && dx == 0) continue;
      int ny = y + dy, nx = x + dx;
      if (ny < 0 || ny >= WID || nx < 0 || nx >= WID) continue;
      int j = ny * WID + nx;
      float d0 = e0 - Eb[j];
      float d1 = e1 - Eb[NNODE + j];
      float d2 = e2 - Eb[2 * NNODE + j];
      float w = expf(-0.5f * (d0 * d0 + d1 * d1 + d2 * d2));
      deg += w;
      row[j] = -ub * w;
    }
  }
  row[i] = 1.0f + ub * deg;
}

// y (B x 1296 x 3, a raw reinterpret of Y[B,3,36,36]) zero-padded to 16 cols.
__global__ void ypad_kernel(const float* __restrict__ Y, float* __restrict__ Ypad) {
  int idx = blockIdx.x * blockDim.x + threadIdx.x;
  if (idx >= BATCH * NNODE) return;
  int b = idx / NNODE, n = idx % NNODE;
  float* d = Ypad + ((size_t)b * NNODE + n) * 16;
  const float* y = Y + (size_t)b * 3 * NNODE + (size_t)n * 3;
  d[0] = y[0]; d[1] = y[1]; d[2] = y[2];
#pragma unroll
  for (int t = 3; t < 16; ++t) d[t] = 0.0f;
}

// Invert the 16x16 pivot block A[k][k] (Gauss-Jordan in LDS, SPD -> no pivoting)
__global__ void pivotinv_kernel(const float* __restrict__ A, float* __restrict__ Pinv, int k) {
  int b = blockIdx.x;
  int t = threadIdx.x;         // 256 threads
  int r = t >> 4, c = t & 15;
  __shared__ float M[16][17];
  __shared__ float I[16][17];
  M[r][c] = A[(size_t)b * NNODE * NNODE + (size_t)(k * 16 + r) * NNODE + k * 16 + c];
  I[r][c] = (r == c) ? 1.0f : 0.0f;
  for (int p = 0; p < 16; ++p) {
    __syncthreads();
    float invp = 1.0f / M[p][p];
    if (r == p) { M[p][c] *= invp; I[p][c] *= invp; }
    __syncthreads();
    float f = M[r][p];
    if (r != p) { M[r][c] -= f * M[p][c]; I[r][c] -= f * I[p][c]; }
  }
  __syncthreads();
  Pinv[b * 256 + t] = I[r][c];
}

// Save NEGATED block-column k (f32 WMMA has no A-negate, only CNeg, so we
// express C - A*B as C + (-A)*B by pre-negating the column).
__global__ void savecol_kernel(const float* __restrict__ A, float* __restrict__ Colneg, int k) {
  int i = blockIdx.x;          // block row
  int b = blockIdx.y;
  int t = threadIdx.x;         // 256 threads
  int r = t >> 4, c = t & 15;
  Colneg[((size_t)b * NBLK + i) * 256 + t] =
      -A[(size_t)b * NNODE * NNODE + (size_t)(i * 16 + r) * NNODE + k * 16 + c];
}

// Row update: A[k][j] <- P @ A[k][j] (j != k), A[k][k] <- P.  4 waves/block.
__global__ void rowupd_kernel(float* __restrict__ A, const float* __restrict__ Pinv, int k) {
  int j = blockIdx.x * 4 + threadIdx.y;
  int b = blockIdx.y;
  if (j >= NBLK) return;                       // wave-uniform
  const float* P = Pinv + (size_t)b * 256;
  float* Ab = A + (size_t)b * NNODE * NNODE;
  if (j == k) {                                // wave-uniform
    int lane = threadIdx.x;
#pragma unroll
    for (int t = 0; t < 8; ++t) {
      int e = t * 32 + lane;
      Ab[(size_t)(k * 16 + (e >> 4)) * NNODE + k * 16 + (e & 15)] = P[e];
    }
    return;
  }
  float* Bblk = Ab + (size_t)(k * 16) * NNODE + (size_t)j * 16;
  v8f acc = v8f_zero();
  acc = wmma_blk16(P, 16, Bblk, NNODE, acc);
  store_c16(Bblk, NNODE, acc);
}

// Elimination over a 4(i) x 4(j) block tile per workgroup (4 waves):
//   pivot-row tile A_new[k][j0..j0+3] staged in LDS via async copy, Col block
//   kept in registers as WMMA A-fragments and reused across the 4 j-tiles.
//   for i != k:  A[i][j] += Colneg_i @ A_new[k][j]   (j != k)
//                A[i][k]  = Colneg_i @ P             (j == k, C = 0)
__global__ void elim_kernel(float* __restrict__ A, const float* __restrict__ Colneg, int k) {
  __shared__ float ldsRow[4][16][16];
  const int b  = blockIdx.z;
  const int i0 = blockIdx.y * 4;
  const int j0 = blockIdx.x * 4;
  float* Ab = A + (size_t)b * NNODE * NNODE;
  const int tid = threadIdx.y * 32 + threadIdx.x;   // 0..127

  // Cooperative stage of the pivot-row tile: 4 blocks * 256 floats = 4 KB,
  // 256 16-byte chunks, 2 per thread.
#pragma unroll
  for (int c = 0; c < 2; ++c) {
    int chunk = tid * 2 + c;          // 0..255
    int jj    = chunk >> 6;           // which j block
    int row   = (chunk >> 2) & 15;    // row within block
    int quad  = chunk & 3;            // 4-float group within row
    int j = j0 + jj;
    if (j < NBLK) {
      const float* src = Ab + (size_t)(k * 16 + row) * NNODE + (size_t)j * 16 + quad * 4;
      float* dst = &ldsRow[jj][row][quad * 4];
#if USE_ASYNC_LDS
      __builtin_amdgcn_global_load_async_to_lds_b128(
          (gv4i_ptr)const_cast<float*>(src), (lv4i_ptr)dst, 0, 0);
#else
      dst[0] = src[0]; dst[1] = src[1]; dst[2] = src[2]; dst[3] = src[3];
#endif
    }
  }
#if USE_ASYNC_LDS
  __builtin_amdgcn_s_wait_asynccnt(0);
#endif
  __syncthreads();

  const int i = i0 + threadIdx.y;     // one i block per wave
  if (i >= NBLK || i == k) return;    // wave-uniform
  AFrag col = load_afrag(Colneg + ((size_t)b * NBLK + i) * 256, 16);
  for (int jj = 0; jj < 4; ++jj) {
    int j = j0 + jj;
    if (j >= NBLK) break;             // wave-uniform
    float* Cij = Ab + (size_t)(i * 16) * NNODE + (size_t)j * 16;
    v8f acc = (j == k) ? v8f_zero() : load_c16(Cij, NNODE);
    acc = wmma_afrag(col, &ldsRow[jj][0][0], 16, acc);
    store_c16(Cij, NNODE, acc);
  }
}

// out[b] (1296x3, raw layout) = A^{-1}[b] @ Ypad[b][:, :3]
__global__ void finalmm_kernel(const float* __restrict__ A, const float* __restrict__ Ypad,
                               float* __restrict__ out) {
  int i = blockIdx.x * 4 + threadIdx.y;        // row block
  int b = blockIdx.y;
  if (i >= NBLK) return;                       // wave-uniform
  const float* Ab = A + (size_t)b * NNODE * NNODE + (size_t)(i * 16) * NNODE;
  const float* Yb = Ypad + (size_t)b * NNODE * 16;
  v8f acc = v8f_zero();
  for (int kb = 0; kb < NBLK; ++kb) {
    if (kb + 1 < NBLK)
      __builtin_prefetch(Ab + (kb + 1) * 16, 0, 1);  // global_prefetch_b8
    acc = wmma_blk16(Ab + (size_t)kb * 16, NNODE, Yb + (size_t)kb * 16 * 16, 16, acc);
  }
  const int lane = threadIdx.x & 31;
  const int half = lane >> 4;
  const int col  = lane & 15;
  if (col < 3) {
#pragma unroll
    for (int r = 0; r < 8; ++r)
      out[(size_t)b * NNODE * 3 + (size_t)(i * 16 + r + 8 * half) * 3 + col] = acc[r];
  }
}

// ---------------------------- Launcher -------------------------------------

extern "C" void kernel_launch(void* const* d_in, const int* in_sizes, int n_in,
                              void* d_out, int out_size, void* d_ws, size_t ws_size,
                              hipStream_t stream) {
  (void)in_sizes; (void)n_in; (void)out_size; (void)ws_size;
  const float* xf = (const float*)d_in[0];

  // param order = setup_inputs() insertion order, (weight, bias) per layer
  enum { F1A, F1B, F2A, F2B, F3A, F3B, FD1, F4A, F4B, FD2, F5A, F5B, F5C,
         Y1, Y2, Y3, Y4, U1, U2, U3, U4, UFC1, UFC2 };
  auto pw = [&](int i) { return (const float*)d_in[1 + 2 * i]; };
  auto pb = [&](int i) { return (const float*)d_in[2 + 2 * i]; };

  // ---- workspace carve-up (floats) ----
  float* ws = (float*)d_ws;
  size_t off = 0;
  auto alloc = [&](size_t n) {
    float* p = ws + off;
    off += (n + 255) & ~(size_t)255;
    return p;
  };
  const size_t ACT = (size_t)BATCH * 64 * WID * WID;  // 1,327,104 floats
  float* buf0 = alloc(ACT);
  float* buf1 = alloc(ACT);
  float* buf2 = alloc(ACT);
  float* buf3 = alloc(ACT);
  float* buf4 = alloc(ACT);
  float* buf5 = alloc(ACT);
  float* Ebuf = alloc((size_t)BATCH * 3 * NNODE);
  float* Ybuf = alloc((size_t)BATCH * 3 * NNODE);
  float* ubuf = alloc(BATCH);
  float* Abuf = alloc((size_t)BATCH * NNODE * NNODE);   // 107.5 MB
  float* Pinv = alloc((size_t)BATCH * 256);
  float* Coln = alloc((size_t)BATCH * NBLK * 256);
  float* Ypad = alloc((size_t)BATCH * NNODE * 16);

  auto conv = [&](const float* in, int pi, float* out, int Cin, int H, int W,
                  int Cout, int stride, int act, int ctot, int coff) {
    int Ho = (H - 1) / stride + 1;
    int Wo = (W - 1) / stride + 1;
    int total = BATCH * Cout * Ho * Wo;
    conv3x3_kernel<<<(total + 255) / 256, 256, 0, stream>>>(
        in, pw(pi), pb(pi), out, Cin, H, W, Cout, stride, Ho, Wo, act, ctot, coff);
  };
  auto upconv = [&](const float* in, int pi, float* out, int Cin, int H, int W,
                    int Cout, int ctot, int coff) {
    int total = BATCH * Cout * (2 * H) * (2 * W);
    upconv_kernel<<<(total + 255) / 256, 256, 0, stream>>>(
        in, pw(pi), pb(pi), out, Cin, H, W, Cout, ctot, coff);
  };
  auto maxpool = [&](const float* in, float* out, int C, int H, int W) {
    int total = BATCH * C * ((H + 1) / 2) * ((W + 1) / 2);
    maxpool_kernel<<<(total + 255) / 256, 256, 0, stream>>>(in, out, C, H, W);
  };
  auto copych = [&](const float* src, float* dst, int C, int HW, int ctot, int coff) {
    int total = BATCH * C * HW;
    copych_kernel<<<(total + 255) / 256, 256, 0, stream>>>(src, dst, C, HW, ctot, coff);
  };

  // ---- cnnf -> E ----
  conv(xf,   F1A, buf0, 3,  36, 36, 32, 1, ACT_LRELU, 32, 0);
  conv(buf0, F1B, buf1, 32, 36, 36, 32, 1, ACT_LRELU, 32, 0);     // o1
  conv(buf1, F2A, buf0, 32, 36, 36, 64, 1, ACT_RELU,  64, 0);
  maxpool(buf0, buf2, 64, 36, 36);                                // [64,18,18]
  conv(buf2, F2B, buf3, 64, 18, 18, 64, 1, ACT_RELU,  64, 0);     // o2
  conv(buf3, F3A, buf2, 64, 18, 18, 64, 1, ACT_RELU,  64, 0);
  maxpool(buf2, buf0, 64, 18, 18);                                // [64,9,9]
  conv(buf0, F3B, buf2, 64, 9, 9, 128, 1, ACT_RELU, 128, 0);      // [128,9,9]
  upconv(buf2, FD1, buf4, 128, 9, 9, 64, 128, 0);                 // cat ch 0-63
  copych(buf3, buf4, 64, 18 * 18, 128, 64);                       // cat ch 64-127
  conv(buf4, F4A, buf0, 128, 18, 18, 64, 1, ACT_LRELU, 64, 0);
  conv(buf0, F4B, buf2, 64, 18, 18, 64, 1, ACT_LRELU, 64, 0);
  upconv(buf2, FD2, buf5, 64, 18, 18, 32, 64, 0);                 // cat2 ch 0-31
  copych(buf1, buf5, 32, 36 * 36, 64, 32);                        // cat2 ch 32-63
  conv(buf5, F5A, buf0, 64, 36, 36, 32, 1, ACT_LRELU, 32, 0);
  conv(buf0, F5B, buf2, 32, 36, 36, 32, 1, ACT_LRELU, 32, 0);
  conv(buf2, F5C, Ebuf, 32, 36, 36, 3, 1, ACT_NONE, 3, 0);        // E

  // ---- cnny -> Y = x + o ----
  conv(xf,   Y1, buf0, 3,  36, 36, 32, 1, ACT_RELU, 32, 0);
  conv(buf0, Y2, buf2, 32, 36, 36, 32, 1, ACT_RELU, 32, 0);
  conv(buf2, Y3, buf0, 32, 36, 36, 32, 1, ACT_RELU, 32, 0);
  conv(buf0, Y4, buf2, 32, 36, 36, 3, 1, ACT_NONE, 3, 0);
  {
    int n = BATCH * 3 * NNODE;
    resadd_kernel<<<(n + 255) / 256, 256, 0, stream>>>(xf, buf2, Ybuf, n);
  }

  // ---- cnnu -> u (clipped) ----
  conv(xf,   U1, buf0, 3,  36, 36, 32, 2, ACT_LRELU, 32, 0);      // [32,18,18]
  conv(buf0, U2, buf2, 32, 18, 18, 32, 1, ACT_LRELU, 32, 0);
  maxpool(buf2, buf0, 32, 18, 18);                                // [32,9,9]
  conv(buf0, U3, buf2, 32, 9, 9, 32, 1, ACT_LRELU, 32, 0);
  maxpool(buf2, buf0, 32, 9, 9);                                  // [32,5,5]
  conv(buf0, U4, buf2, 32, 5, 5, 32, 1, ACT_LRELU, 32, 0);
  maxpool(buf2, buf0, 32, 5, 5);                                  // [32,3,3]
  cnnu_fc_kernel<<<BATCH, 32, 0, stream>>>(buf0, pw(UFC1), pb(UFC1),
                                           pw(UFC2), pb(UFC2), ubuf);

  // ---- A = I + u L ----
  {
    long long nA = (long long)BATCH * NNODE * NNODE;
    zero_kernel<<<4096, 256, 0, stream>>>(Abuf, nA);
    int n = BATCH * NNODE;
    buildA_kernel<<<(n + 255) / 256, 256, 0, stream>>>(Ebuf, ubuf, Abuf);
    ypad_kernel<<<(n + 255) / 256, 256, 0, stream>>>(Ybuf, Ypad);
  }

  // ---- blocked in-place Gauss-Jordan inversion (WMMA f32 16x16x4) ----
  const int TG = (NBLK + 3) / 4;  // 21 tiles of 4 blocks
  for (int k = 0; k < NBLK; ++k) {
    pivotinv_kernel<<<BATCH, 256, 0, stream>>>(Abuf, Pinv, k);
    savecol_kernel<<<dim3(NBLK, BATCH), 256, 0, stream>>>(Abuf, Coln, k);
    rowupd_kernel<<<dim3(TG, BATCH), dim3(32, 4), 0, stream>>>(Abuf, Pinv, k);
    elim_kernel<<<dim3(TG, TG, BATCH), dim3(32, 4), 0, stream>>>(Abuf, Coln, k);
  }

  // ---- out = A^{-1} @ y ----
  finalmm_kernel<<<dim3(TG, BATCH), dim3(32, 4), 0, stream>>>(Abuf, Ypad,
                                                              (float*)d_out);
}